// ConditionalExpertRouting_82652350644290
// MI455X (gfx1250) — compile-verified
//
#include <hip/hip_runtime.h>
#include <math.h>

// Problem constants (from reference)
#define B_ 4
#define S_ 2048
#define H_ 512
#define E_ 8
#define P_ 4
#define I_ 2048
#define R_ (B_ * S_)          // 8192 token rows
#define KC 128                // I-dimension chunk per fused step
#define XPAD 516              // LDS pitch for x tile (516 % 64 == 4 -> conflict-free)
#define HPAD 132              // LDS pitch for hidden tile

typedef __attribute__((ext_vector_type(2))) float v2f;
typedef __attribute__((ext_vector_type(8))) float v8f;

// D = A(16x4) * B(4x16) + C, fp32 matrix core (gfx1250 wave32 WMMA)
__device__ __forceinline__ v8f wmma_f32(v2f a, v2f b, v8f c) {
  return __builtin_amdgcn_wmma_f32_16x16x4_f32(
      /*neg_a=*/false, a, /*neg_b=*/false, b,
      /*c_mod=*/(short)0, c, /*reuse_a=*/false, /*reuse_b=*/false);
}

// ---------------------------------------------------------------------------
// Kernel 1: router. One wave32 per token row.
// ew layout: ew[e * R_ + r]  (expert-major so kernel 2 reads contiguously)
// ---------------------------------------------------------------------------
__global__ __launch_bounds__(256)
void moe_router_kernel(const float* __restrict__ x,
                       const float* __restrict__ Wp,
                       const float* __restrict__ bp,
                       const float* __restrict__ Ws,
                       const float* __restrict__ bs,
                       float* __restrict__ ew) {
  const int gwave = (int)((blockIdx.x * blockDim.x + threadIdx.x) >> 5);
  const int lane  = (int)(threadIdx.x & 31);
  if (gwave >= R_) return;

  const float* xr = x + (size_t)gwave * H_;
  float a0 = 0.f, a1 = 0.f, a2 = 0.f, a3 = 0.f;
#pragma unroll 4
  for (int j = 0; j < H_ / 32; ++j) {
    const int h = lane + j * 32;          // coalesced across lanes
    const float xv = xr[h];
    a0 += xv * Wp[0 * H_ + h];
    a1 += xv * Wp[1 * H_ + h];
    a2 += xv * Wp[2 * H_ + h];
    a3 += xv * Wp[3 * H_ + h];
  }
#pragma unroll
  for (int off = 16; off >= 1; off >>= 1) {
    a0 += __shfl_xor(a0, off, 32);
    a1 += __shfl_xor(a1, off, 32);
    a2 += __shfl_xor(a2, off, 32);
    a3 += __shfl_xor(a3, off, 32);
  }
  if (lane == 0) {
    float p[4] = {a0 + bp[0], a1 + bp[1], a2 + bp[2], a3 + bp[3]};
    float pmax = fmaxf(fmaxf(p[0], p[1]), fmaxf(p[2], p[3]));
    float psum = 0.f;
#pragma unroll
    for (int i = 0; i < 4; ++i) { p[i] = __expf(p[i] - pmax); psum += p[i]; }
    const float pinv = 1.f / psum;
#pragma unroll
    for (int i = 0; i < 4; ++i) p[i] *= pinv;

    float el[E_];
    float emax = -1e30f;
#pragma unroll
    for (int e = 0; e < E_; ++e) {
      float v = bs[e];
#pragma unroll
      for (int q = 0; q < 4; ++q) v += p[q] * Ws[e * P_ + q];
      el[e] = v;
      emax = fmaxf(emax, v);
    }
    float esum = 0.f;
#pragma unroll
    for (int e = 0; e < E_; ++e) { el[e] = __expf(el[e] - emax); esum += el[e]; }
    const float einv = 1.f / esum;
#pragma unroll
    for (int e = 0; e < E_; ++e) ew[(size_t)e * R_ + gwave] = el[e] * einv;
  }
}

// ---------------------------------------------------------------------------
// Kernel 2: fused dense-MoE MLP + weighted combine.
// Grid: 512 workgroups (16 token rows each), 256 threads = 8 waves.
// Wave w owns output columns [w*64, w*64+64) of H; 4 WMMA C-tiles.
// hidden [B,S,E,I] is never materialized (saves ~1 GB of HBM traffic).
// ---------------------------------------------------------------------------
__global__ __launch_bounds__(256)
void moe_fused_kernel(const float* __restrict__ x,
                      const float* __restrict__ Wup,
                      const float* __restrict__ Wdn,
                      const float* __restrict__ ew,
                      float* __restrict__ out) {
  __shared__ float x_lds[16][XPAD];     // 16 x 512 x-tile, padded pitch
  __shared__ float hid_lds[16][HPAD];   // 16 x 128 hidden chunk, padded pitch
  __shared__ float w_lds[E_][16];       // router weights for this row tile

  const int row0   = (int)blockIdx.x * 16;
  const int tid    = (int)threadIdx.x;
  const int wv     = tid >> 5;          // wave id 0..7
  const int lane   = tid & 31;
  const int lane16 = lane & 15;
  const int khalf  = (lane >> 4) * 2;   // K sub-offset for A/B operands: 0 or 2
  const int hi8    = (lane >> 4) * 8;   // C/D row offset: 0 or 8

  // Stage x tile (coalesced) and router weights.
  for (int idx = tid; idx < 16 * H_; idx += 256) {
    const int r = idx >> 9;             // / 512
    const int c = idx & (H_ - 1);
    x_lds[r][c] = x[(size_t)(row0 + r) * H_ + c];
  }
  if (tid < E_ * 16) {
    const int e = tid >> 4, rr = tid & 15;
    w_lds[e][rr] = ew[(size_t)e * R_ + row0 + rr];
  }
  __syncthreads();

  const v8f vzero = {};
  v8f fin[4];
#pragma unroll
  for (int t = 0; t < 4; ++t) fin[t] = vzero;

  for (int e = 0; e < E_; ++e) {
    const float* Wup_e = Wup + (size_t)e * I_ * H_;
    const float* Wdn_e = Wdn + (size_t)e * H_ * I_;

    v8f eacc[4];
#pragma unroll
    for (int t = 0; t < 4; ++t) eacc[t] = vzero;

    for (int i0 = 0; i0 < I_; i0 += KC) {
      // ---- GEMM1: this wave computes hidden[:, i0 + wv*16 + (0..15)] ----
      // B operand: lane l supplies W_up[e][i0 + wv*16 + l%16][k0 + khalf + {0,1}]
      v8f hacc = vzero;
      const float* bptr = Wup_e + (size_t)(i0 + wv * 16 + lane16) * H_ + khalf;
#pragma unroll 4
      for (int k0 = 0; k0 < H_; k0 += 4) {
        v2f a, b;
        a[0] = x_lds[lane16][k0 + khalf];
        a[1] = x_lds[lane16][k0 + khalf + 1];
        b[0] = bptr[k0];
        b[1] = bptr[k0 + 1];
        hacc = wmma_f32(a, b, hacc);
      }
      // exact GELU, stage hidden chunk in LDS
#pragma unroll
      for (int v = 0; v < 8; ++v) {
        const float hv = hacc[v];
        const float g = 0.5f * hv * (1.0f + erff(hv * 0.70710678118654752f));
        hid_lds[v + hi8][wv * 16 + lane16] = g;
      }
      __syncthreads();

      // ---- GEMM2: eacc[16 x 64] += hidden[16 x 128] * W_down[n, i]^T ----
#pragma unroll
      for (int t = 0; t < 4; ++t) {
        const int n = wv * 64 + t * 16 + lane16;
        const float* b2 = Wdn_e + (size_t)n * I_ + i0 + khalf;
#pragma unroll 4
        for (int k0 = 0; k0 < KC; k0 += 4) {
          v2f a, b;
          a[0] = hid_lds[lane16][k0 + khalf];
          a[1] = hid_lds[lane16][k0 + khalf + 1];
          b[0] = b2[k0];
          b[1] = b2[k0 + 1];
          eacc[t] = wmma_f32(a, b, eacc[t]);
        }
      }
      __syncthreads();   // before next chunk overwrites hid_lds
    }

    // ---- softmax-weighted combine over experts (register-resident) ----
#pragma unroll
    for (int t = 0; t < 4; ++t) {
#pragma unroll
      for (int v = 0; v < 8; ++v) {
        const float wgt = w_lds[e][v + hi8];   // LDS broadcast per half-wave
        fin[t][v] += wgt * eacc[t][v];
      }
    }
  }

  // ---- store final [16 x 512] tile ----
#pragma unroll
  for (int t = 0; t < 4; ++t) {
    const int col = wv * 64 + t * 16 + lane16;
#pragma unroll
    for (int v = 0; v < 8; ++v) {
      const int row = row0 + v + hi8;
      out[(size_t)row * H_ + col] = fin[t][v];
    }
  }
}

// ---------------------------------------------------------------------------
extern "C" void kernel_launch(void* const* d_in, const int* in_sizes, int n_in,
                              void* d_out, int out_size, void* d_ws, size_t ws_size,
                              hipStream_t stream) {
  const float* x    = (const float*)d_in[0];
  const float* Wp   = (const float*)d_in[1];
  const float* bp   = (const float*)d_in[2];
  const float* Ws   = (const float*)d_in[3];
  const float* bs   = (const float*)d_in[4];
  const float* Wup  = (const float*)d_in[5];
  const float* Wdn  = (const float*)d_in[6];
  float* out = (float*)d_out;
  float* ew  = (float*)d_ws;            // [E_][R_] router weights, 256 KB

  // Router: one wave per token row -> 8192 waves -> 1024 blocks of 8 waves.
  moe_router_kernel<<<R_ / 8, 256, 0, stream>>>(x, Wp, bp, Ws, bs, ew);

  // Fused expert MLP + combine: 512 row-tiles of 16 rows.
  moe_fused_kernel<<<R_ / 16, 256, 0, stream>>>(x, Wup, Wdn, ew, out);
}